// NFBottleneck_71932112273618
// MI455X (gfx1250) — compile-verified
//
#include <hip/hip_runtime.h>
#include <math.h>

// Fused planar normalizing flow for MI455X (gfx1250, wave32).
//
//   lin_k = z0.w_k + b_k + sum_{j<k} t_j (uhat_j . w_k)
//   z_out = z0 + T @ Uhat          (T = tanh of the lin scan)
//   logdet_k = log(|1 + (1-t_k^2) * (w_k.uhat_k)| + 1e-8)
//
// GEMM1 (B x 256 x 32) and GEMM2 (B x 32 x 256) use V_WMMA_F32_16X16X4_F32.
// Kernel is HBM-bandwidth bound (~136 MB mandatory traffic ~= 6 us @ 23.3 TB/s);
// the epilogue re-read of X hits the 192 MB L2.

typedef float v2f __attribute__((ext_vector_type(2)));
typedef float v8f __attribute__((ext_vector_type(8)));

#define DD        256   // latent dim
#define KK        32    // flow depth
#define NWAVES    4     // waves per block
#define TILE_ROWS 16    // rows per wave (WMMA M)
#define STR       34    // LDS row stride in words: even (8B-aligned b64) + bank-spread

__global__ __launch_bounds__(128, 1)
void planar_flow_fused(const float* __restrict__ X,
                       const float* __restrict__ U,
                       const float* __restrict__ W,
                       const float* __restrict__ Bv,
                       float* __restrict__ Zout,
                       float* __restrict__ Lout,
                       int Btot)
{
    __shared__ float uhatT[DD * STR];              // uhatT[d*STR + k] = uhat[k][d]
    __shared__ float Ms[KK * STR];                 // Ms[j*STR + k]   = uhat_j . w_k
    __shared__ float bs[KK];
    __shared__ float alphas[KK];
    __shared__ float lin_s[NWAVES][TILE_ROWS * STR];
    __shared__ float T_s[NWAVES][TILE_ROWS * STR];

    const int tid  = threadIdx.x;
    const int wave = tid >> 5;
    const int lane = tid & 31;
    const int n    = lane & 15;   // N (or M-row) index within fragment
    const int h    = lane >> 4;   // K-half selector

    // ---------------- per-block setup (all data L2-hot, 64 KB total) ----------
    if (tid < KK) {
        const float* wk = W + tid * DD;
        const float* uk = U + tid * DD;
        float wu = 0.f, ww = 0.f;
        for (int d = 0; d < DD; ++d) {
            float wv = wk[d];
            wu = fmaf(wv, uk[d], wu);
            ww = fmaf(wv, wv, ww);
        }
        float sp = logf(1.f + expf(wu));            // softplus(wu)
        alphas[tid] = (sp - 1.f - wu) / ww;
        bs[tid]     = Bv[tid];
    }
    __syncthreads();

    // uhat transposed: contiguous k per d-row -> b64 B-fragment loads in GEMM2
    for (int i = tid; i < KK * DD; i += 128) {
        int k = i >> 8, d = i & (DD - 1);
        uhatT[d * STR + k] = U[i] + alphas[k] * W[i];
    }
    __syncthreads();

    // M[j][k] = uhat_j . w_k   (includes diagonal c_k = M[k][k])
    for (int i = tid; i < KK * KK; i += 128) {
        int j = i >> 5, k = i & 31;
        const float* wk = W + k * DD;
        float s = 0.f;
        for (int d = 0; d < DD; ++d) s = fmaf(uhatT[d * STR + j], wk[d], s);
        Ms[j * STR + k] = s;
    }
    __syncthreads();

    // ---------------- GEMM1: L0 = z0 @ W^T  (16 x 256 x 32 per wave) ----------
    const int row0 = (blockIdx.x * NWAVES + wave) * TILE_ROWS;

    v8f acc0 = {};   // flows 0..15
    v8f acc1 = {};   // flows 16..31
    const float* Arow  = X + (size_t)(row0 + n) * DD + h * 2;   // A: z0[row=n][kc+2h+{0,1}]
    const float* Brow0 = W + (size_t)n        * DD + h * 2;     // B: w[col=n   ][kc+2h+{0,1}]
    const float* Brow1 = W + (size_t)(n + 16) * DD + h * 2;     // B: w[col=n+16][...]
    #pragma unroll 8
    for (int kc = 0; kc < DD; kc += 4) {
        v2f a  = *(const v2f*)(Arow  + kc);
        v2f b0 = *(const v2f*)(Brow0 + kc);
        v2f b1 = *(const v2f*)(Brow1 + kc);
        acc0 = __builtin_amdgcn_wmma_f32_16x16x4_f32(false, a, false, b0, (short)0, acc0, false, false);
        acc1 = __builtin_amdgcn_wmma_f32_16x16x4_f32(false, a, false, b1, (short)0, acc1, false, false);
    }

    // C-layout (v,lane) -> (row = v + 8h, col = n | 16+n): spill lin to LDS
    float* linW = &lin_s[wave][0];
    #pragma unroll
    for (int v = 0; v < 8; ++v) {
        int r = v + 8 * h;
        linW[r * STR + n]      = acc0[v];
        linW[r * STR + 16 + n] = acc1[v];
    }
    __syncthreads();

    // ---------------- triangular scan: lanes 0..15 own one row each -----------
    if (h == 0) {
        const int r = n;
        float lin[KK];
        #pragma unroll
        for (int k = 0; k < KK; ++k) lin[k] = linW[r * STR + k] + bs[k];

        float* Tw = &T_s[wave][0];
        #pragma unroll
        for (int k = 0; k < KK; ++k) {
            float tk = tanhf(lin[k]);
            Tw[r * STR + k] = tk;
            float ck = Ms[k * STR + k];
            Lout[(size_t)k * Btot + row0 + r] =
                logf(fabsf(1.f + (1.f - tk * tk) * ck) + 1e-8f);
            #pragma unroll
            for (int k2 = k + 1; k2 < KK; ++k2)
                lin[k2] = fmaf(tk, Ms[k * STR + k2], lin[k2]);
        }
    }
    __syncthreads();

    // ---------------- GEMM2: z = z0 + T @ Uhat  (16 x 32 x 256 per wave) ------
    const float* Tw = &T_s[wave][0];
    for (int tt = 0; tt < 16; ++tt) {            // 16 N-tiles covering D=256
        v8f acc = {};
        #pragma unroll
        for (int kc = 0; kc < KK; kc += 4) {
            v2f a  = *(const v2f*)(Tw + n * STR + kc + h * 2);                 // A: T[row=n][k]
            v2f bb = *(const v2f*)(&uhatT[(tt * 16 + n) * STR + kc + h * 2]);  // B: uhat[k][col]
            acc = __builtin_amdgcn_wmma_f32_16x16x4_f32(false, a, false, bb, (short)0, acc, false, false);
        }
        #pragma unroll
        for (int v = 0; v < 8; ++v) {
            size_t idx = (size_t)(row0 + v + 8 * h) * DD + tt * 16 + n;
            Zout[idx] = acc[v] + X[idx];          // X re-read is L2-resident
        }
    }
}

extern "C" void kernel_launch(void* const* d_in, const int* in_sizes, int n_in,
                              void* d_out, int out_size, void* d_ws, size_t ws_size,
                              hipStream_t stream)
{
    // setup_inputs order: X, h(unused), u, w, b
    const float* X  = (const float*)d_in[0];
    const float* U  = (const float*)d_in[2];
    const float* W  = (const float*)d_in[3];
    const float* Bv = (const float*)d_in[4];

    const int Btot = in_sizes[0] / DD;            // 65536
    float* Zout = (float*)d_out;                  // (B, D)
    float* Lout = (float*)d_out + (size_t)Btot * DD;  // (K, B) appended

    const int blocks = Btot / (NWAVES * TILE_ROWS);   // 1024
    planar_flow_fused<<<blocks, 128, 0, stream>>>(X, U, W, Bv, Zout, Lout, Btot);
}